// ModelTorch_28681791602766
// MI455X (gfx1250) — compile-verified
//
#include <hip/hip_runtime.h>
#include <stdint.h>

// Stream compaction: mask -> exclusive prefix count -> gather -> scatter.
// Memory-bound (0 FLOPs -> no WMMA applies). MI455X reasoning:
//  - HBM floor ~224MB / 23.3 TB/s ~= 10us.
//  - 128MB gather table kept L2-resident (192MB L2) by making all output
//    stores nontemporal (TH=NT) so write traffic doesn't evict it.
//  - idx (32MB) read in pass1 stays in L2 for the pass3 re-read.
//  - CDNA5 async global->LDS DMA (ASYNCcnt) stages index tiles in pass3.
//  - wave32-native __shfl_up scans (5 steps) instead of LDS Hillis-Steele.

#define TILE 4096          // int32 elements per block tile (16 KB LDS)
#define BLOCK 256          // 8 wave32 waves per block
#define PER_THREAD (TILE / BLOCK)  // 16
#define NWAVES (BLOCK / 32)        // 8

// Native clang vector types (HIP float4/int4 are structs; builtins want these).
typedef float v4f __attribute__((ext_vector_type(4)));
typedef int   v4i __attribute__((ext_vector_type(4)));

// -------- address-space cast helpers (integer round-trip; flat->LDS is a
// low-32-bit truncation per CDNA5 ISA 10.2, flat->global is identity).
// The async builtins take int4-typed AS(1)/AS(3) pointers. --------
typedef __attribute__((address_space(1))) v4i* glb_v4i_t;
typedef __attribute__((address_space(3))) v4i* lds_v4i_t;

__device__ __forceinline__ lds_v4i_t to_lds4(void* p) {
    return (lds_v4i_t)(uint32_t)(uintptr_t)p;
}
__device__ __forceinline__ glb_v4i_t to_glb4(const void* p) {
    return (glb_v4i_t)(uintptr_t)p;
}

#if __has_builtin(__builtin_amdgcn_global_load_async_to_lds_b128)
#define HAVE_ASYNC_LDS 1
#else
#define HAVE_ASYNC_LDS 0
#endif

__device__ __forceinline__ void wait_asynccnt0() {
#if __has_builtin(__builtin_amdgcn_s_wait_asynccnt)
    __builtin_amdgcn_s_wait_asynccnt(0);
#else
    asm volatile("s_wait_asynccnt 0" ::: "memory");
#endif
}

// Inclusive scan of `v` across the 32 lanes of a wave (5 shfl steps).
__device__ __forceinline__ int wave_incl_scan(int v, int lane) {
#pragma unroll
    for (int d = 1; d < 32; d <<= 1) {
        int y = __shfl_up(v, d, 32);
        if (lane >= d) v += y;
    }
    return v;
}

// -------- pass 0: zero the output (scatter only fills [0, total)) --------
__global__ void compact_zero_kernel(float* __restrict__ out, int n) {
    int i = (blockIdx.x * blockDim.x + threadIdx.x) * 4;
    if (i + 3 < n) {
        v4f z = {0.f, 0.f, 0.f, 0.f};
        __builtin_nontemporal_store(z, (v4f*)(out + i));      // b128 TH=NT
    } else {
        for (int j = i; j < n; ++j) __builtin_nontemporal_store(0.f, out + j);
    }
}

// -------- pass 1: per-block valid counts --------
__global__ void compact_count_kernel(const int* __restrict__ idx,
                                     int* __restrict__ blockSums, int n) {
    __shared__ int wsum[NWAVES];
    const int tid  = threadIdx.x;
    const int lane = tid & 31;
    const int wid  = tid >> 5;
    const int base = blockIdx.x * TILE;
    int cnt = 0;
    if (base + TILE <= n) {
        const v4i* p = (const v4i*)(idx + base);
#pragma unroll
        for (int k = 0; k < PER_THREAD / 4; ++k) {
            v4i v = p[k * BLOCK + tid];            // coalesced b128 loads
            cnt += (v.x >= 0) + (v.y >= 0) + (v.z >= 0) + (v.w >= 0);
        }
    } else {
        for (int j = base + tid; j < n; j += BLOCK) cnt += (idx[j] >= 0);
    }
    // wave32 reduce, then 8 wave sums in LDS
    int ws = wave_incl_scan(cnt, lane);            // lane31 holds wave total
    if (lane == 31) wsum[wid] = ws;
    __syncthreads();
    if (tid == 0) {
        int acc = 0;
#pragma unroll
        for (int w = 0; w < NWAVES; ++w) acc += wsum[w];
        blockSums[blockIdx.x] = acc;
    }
}

// -------- pass 2: single-block exclusive scan of block sums (nb <= 4096) ----
__global__ void compact_scan_kernel(const int* __restrict__ sums,
                                    int* __restrict__ offs, int nb) {
    __shared__ int s[1024];
    const int t = threadIdx.x;
    int v[4];
    int loc = 0;
#pragma unroll
    for (int j = 0; j < 4; ++j) {
        int i = t * 4 + j;
        v[j] = (i < nb) ? sums[i] : 0;
        loc += v[j];
    }
    s[t] = loc;
    __syncthreads();
    for (int off = 1; off < 1024; off <<= 1) {   // Hillis-Steele inclusive
        int x = (t >= off) ? s[t - off] : 0;
        __syncthreads();
        s[t] += x;
        __syncthreads();
    }
    int running = s[t] - loc;                    // exclusive for this thread
#pragma unroll
    for (int j = 0; j < 4; ++j) {
        int i = t * 4 + j;
        if (i < nb) offs[i] = running;
        running += v[j];
    }
}

// -------- pass 3: gather + ordered scatter; tile staged via async DMA -------
__global__ void compact_scatter_kernel(const int* __restrict__ idx,
                                       const float* __restrict__ cache,
                                       const int* __restrict__ blockOffs,
                                       float* __restrict__ out, int n) {
    __shared__ int tile[TILE];                   // 16 KB
    __shared__ int wsum[NWAVES];
    const int tid  = threadIdx.x;
    const int lane = tid & 31;
    const int wid  = tid >> 5;
    const int base = blockIdx.x * TILE;

    if (base + TILE <= n) {
#if HAVE_ASYNC_LDS
        // CDNA5 async global->LDS DMA: 4x b128 per lane, ASYNCcnt-tracked.
        const char* g = (const char*)(idx + base);
        char*       l = (char*)tile;
#pragma unroll
        for (int k = 0; k < 4; ++k) {
            int c = (k * BLOCK + tid) * 16;      // 16B chunk, coalesced
            __builtin_amdgcn_global_load_async_to_lds_b128(
                to_glb4(g + c), to_lds4(l + c), 0, 0);
        }
        wait_asynccnt0();                        // this wave's DMAs landed
#else
        const v4i* p = (const v4i*)(idx + base);
        v4i*      tp = (v4i*)tile;
#pragma unroll
        for (int k = 0; k < 4; ++k) tp[k * BLOCK + tid] = p[k * BLOCK + tid];
#endif
    } else {
        for (int j = tid; j < TILE; j += BLOCK)
            tile[j] = (base + j < n) ? idx[base + j] : -1;
    }
    __syncthreads();                             // LDS visible to all 8 waves

    // Each thread owns PER_THREAD contiguous elements -> rank order preserved.
    int my[PER_THREAD];
    int cnt = 0;
#pragma unroll
    for (int j = 0; j < PER_THREAD; ++j) {
        my[j] = tile[tid * PER_THREAD + j];
        cnt += (my[j] >= 0);
    }
    // Block-exclusive prefix of per-thread counts: wave scan + wave-sum scan.
    int inc = wave_incl_scan(cnt, lane);
    if (lane == 31) wsum[wid] = inc;
    __syncthreads();
    if (tid == 0) {
        int acc = 0;
#pragma unroll
        for (int w = 0; w < NWAVES; ++w) { int t2 = wsum[w]; wsum[w] = acc; acc += t2; }
    }
    __syncthreads();
    int dst = blockOffs[blockIdx.x] + wsum[wid] + (inc - cnt);
#pragma unroll
    for (int j = 0; j < PER_THREAD; ++j) {
        int ix = my[j];
        if (ix >= 0) {
            float v = cache[ix];                          // L2-resident gather
            __builtin_nontemporal_store(v, out + dst++);  // TH=NT, spare L2
        }
    }
}

extern "C" void kernel_launch(void* const* d_in, const int* in_sizes, int n_in,
                              void* d_out, int out_size, void* d_ws, size_t ws_size,
                              hipStream_t stream) {
    const int*   accept = (const int*)d_in[0];     // accept_index
    const float* cache  = (const float*)d_in[1];   // out_cache_loc
    float*       out    = (float*)d_out;
    const int n  = in_sizes[0];
    const int nb = (n + TILE - 1) / TILE;          // 2048 for 8.4M

    int* sums = (int*)d_ws;
    int* offs = sums + nb;

    int zb = ((out_size + 3) / 4 + BLOCK - 1) / BLOCK;
    compact_zero_kernel<<<zb, BLOCK, 0, stream>>>(out, out_size);
    compact_count_kernel<<<nb, BLOCK, 0, stream>>>(accept, sums, n);
    compact_scan_kernel<<<1, 1024, 0, stream>>>(sums, offs, nb);
    compact_scatter_kernel<<<nb, BLOCK, 0, stream>>>(accept, cache, offs, out, n);
}